// Graph_Learn_26113401159747
// MI455X (gfx1250) — compile-verified
//
#include <hip/hip_runtime.h>
#include <math.h>

typedef __attribute__((ext_vector_type(2))) float v2f;
typedef __attribute__((ext_vector_type(8))) float v8f;

#define NB    8
#define TT    16
#define VV    64
#define FF    128
#define NT    (NB * TT)          // 128 (n,t) slices
#define SLICE (VV * FF)          // 8192 floats per (n,t) x-slice
#define PAIRS (VV * VV)          // 4096
#define XSTR  132                // LDS row stride for x tile (16B aligned, low-conflict)
#define TSTR  65                 // LDS row stride for tmpS tile
#define ALPHA 1e-4f

// ---------------------------------------------------------------------------
// K1: per (n,t) slice -> outputs[n,t,i,j] = 0.7 * softmax-style S + 0.3 * I
//     colsum over i is done with V_WMMA_F32_16X16X4_F32 (ones-matrix trick).
// ---------------------------------------------------------------------------
__global__ __launch_bounds__(256) void k1_outputs(const float* __restrict__ x,
                                                  const float* __restrict__ a,
                                                  float* __restrict__ out) {
    __shared__ float xs[VV * XSTR];   // 33.0 KB
    __shared__ float ts[VV * TSTR];   // 16.6 KB
    __shared__ float as[FF];
    __shared__ float colsum[VV];

    const int  tid   = threadIdx.x;
    const long xbase = (long)blockIdx.x * SLICE;
    const long obase = (long)blockIdx.x * PAIRS;

    // stage x[n,t] (64x128) into LDS, padded rows
    for (int k = tid; k < SLICE; k += 256)
        xs[(k >> 7) * XSTR + (k & 127)] = x[xbase + k];
    if (tid < FF) as[tid] = a[tid];
    __syncthreads();

    // pairwise abs-diff dots against a, then exp
    for (int p = tid; p < PAIRS; p += 256) {
        const int i = p >> 6, j = p & 63;
        const float4* xi = (const float4*)&xs[i * XSTR];
        const float4* xj = (const float4*)&xs[j * XSTR];
        const float4* av = (const float4*)as;
        float dot = 0.0f;
#pragma unroll 4
        for (int fc = 0; fc < FF / 4; ++fc) {
            float4 u = xi[fc], w = xj[fc], c = av[fc];
            dot += fabsf(u.x - w.x) * c.x + fabsf(u.y - w.y) * c.y +
                   fabsf(u.z - w.z) * c.z + fabsf(u.w - w.w) * c.w;
        }
        ts[i * TSTR + j] = expf(dot * (1.0f / (float)FF));
    }
    __syncthreads();

    // colsum[j] = sum_i ts[i][j] via WMMA f32 16x16x4 with A = ones.
    // Waves 0..3 each own a 16-wide j group; branch is wave-uniform -> EXEC all 1s.
    const int wave = tid >> 5, lane = tid & 31;
    if (wave < 4) {
        const int jb = wave * 16;
        const int jj = jb + (lane & 15);
        v8f c = {};
        v2f af;
        af[0] = 1.0f;  // A-matrix of all ones (16x4): every lane holds {1,1}
        af[1] = 1.0f;
        for (int kc = 0; kc < 16; ++kc) {
            // B (4x16): VGPR0 = rows K=0 (lanes 0-15) / K=2 (lanes 16-31),
            //           VGPR1 = rows K=1 / K=3 ; N = lane&15
            const int kr = kc * 4 + ((lane < 16) ? 0 : 2);
            v2f bf;
            bf[0] = ts[kr * TSTR + jj];
            bf[1] = ts[(kr + 1) * TSTR + jj];
            c = __builtin_amdgcn_wmma_f32_16x16x4_f32(
                    false, af, false, bf, (short)0, c, false, false);
        }
        // D row M=0 lives in VGPR0 of lanes 0..15
        if (lane < 16) colsum[jb + lane] = c[0];
    }
    __syncthreads();

    for (int p = tid; p < PAIRS; p += 256) {
        const int i = p >> 6, j = p & 63;
        const float s = ts[i * TSTR + j] / (colsum[j] + 1e-8f);
        out[obase + p] = 0.7f * s + ((i == j) ? 0.3f : 0.0f);
    }
}

// ---------------------------------------------------------------------------
// K2: SMsum[i,j] = sum_t mean_n outputs ; per-block loss2 partials
// ---------------------------------------------------------------------------
__global__ __launch_bounds__(256) void k2_smean(const float* __restrict__ out,
                                                float* __restrict__ smsum,
                                                float* __restrict__ l2part) {
    __shared__ float red[256];
    const int ij = blockIdx.x * 256 + threadIdx.x;  // 16 blocks cover 4096 (i,j)
    float smtot = 0.0f, l2 = 0.0f;
    for (int t = 0; t < TT; ++t) {
        float s = 0.0f;
#pragma unroll
        for (int n = 0; n < NB; ++n)
            s += out[(n * TT + t) * PAIRS + ij];
        s *= (1.0f / (float)NB);
        smtot += s;
        l2 += s * s;
    }
    smsum[ij] = smtot;

    red[threadIdx.x] = l2;
    __syncthreads();
    for (int st = 128; st > 0; st >>= 1) {
        if (threadIdx.x < st) red[threadIdx.x] += red[threadIdx.x + st];
        __syncthreads();
    }
    if (threadIdx.x == 0) l2part[blockIdx.x] = red[0];
}

// ---------------------------------------------------------------------------
// K3: per i-row: lacc = sum_j sum_f (diff_mean[i,j,f])^2 * SMsum[i,j]
// ---------------------------------------------------------------------------
__global__ __launch_bounds__(128) void k3_diff(const float* __restrict__ x,
                                               const float* __restrict__ smsum,
                                               float* __restrict__ l1part) {
    __shared__ float xi[NT * FF];   // 64 KB: x[:, :, i, :]
    __shared__ float red[128];
    const int i = blockIdx.x, f = threadIdx.x;

    for (int k = f; k < NT * FF; k += 128) {
        const int nt = k >> 7, ff = k & 127;
        xi[k] = x[(long)nt * SLICE + i * FF + ff];
    }
    __syncthreads();

    float lacc = 0.0f;
    for (int j = 0; j < VV; ++j) {
        const float* xj = x + j * FF + f;
        float m = 0.0f;
#pragma unroll 4
        for (int nt = 0; nt < NT; ++nt)
            m += fabsf(xi[nt * FF + f] - xj[(long)nt * SLICE]);
        const float md = m * (1.0f / (float)NT);   // diff_mean element
        lacc += md * md * smsum[i * VV + j];
    }

    red[f] = lacc;
    __syncthreads();
    for (int st = 64; st > 0; st >>= 1) {
        if (f < st) red[f] += red[f + st];
        __syncthreads();
    }
    if (f == 0) l1part[i] = red[0];
}

// ---------------------------------------------------------------------------
// K4: deterministic final reduction of partials into the two loss scalars
// ---------------------------------------------------------------------------
__global__ void k4_final(const float* __restrict__ l1part,
                         const float* __restrict__ l2part,
                         float* __restrict__ out) {
    if (threadIdx.x == 0 && blockIdx.x == 0) {
        float l1 = 0.0f;
        for (int b = 0; b < VV; ++b) l1 += l1part[b];
        float l2 = 0.0f;
        for (int b = 0; b < 16; ++b) l2 += l2part[b];
        out[(long)NT * PAIRS + 0] = ALPHA * l1;
        out[(long)NT * PAIRS + 1] = ALPHA * l2;
    }
}

extern "C" void kernel_launch(void* const* d_in, const int* in_sizes, int n_in,
                              void* d_out, int out_size, void* d_ws, size_t ws_size,
                              hipStream_t stream) {
    const float* x = (const float*)d_in[0];   // (8,16,64,128) f32
    const float* a = (const float*)d_in[1];   // (128,1) f32
    float* out = (float*)d_out;               // 524288 outputs + loss1 + loss2

    float* smsum  = (float*)d_ws;             // 4096 floats
    float* l1part = smsum + PAIRS;            // 64 floats
    float* l2part = l1part + VV;              // 16 floats

    k1_outputs<<<NT, 256, 0, stream>>>(x, a, out);
    k2_smean<<<PAIRS / 256, 256, 0, stream>>>(out, smsum, l2part);
    k3_diff<<<VV, 128, 0, stream>>>(x, smsum, l1part);
    k4_final<<<1, 32, 0, stream>>>(l1part, l2part, out);
}